// CrossViewSwapAttention_15564961481523
// MI455X (gfx1250) — compile-verified
//
#include <hip/hip_runtime.h>
#include <hip/hip_bf16.h>
#include <math.h>

// ---------------------------------------------------------------------------
// CDNA5 (gfx1250) implementation of CrossViewSwapAttention.
// All GEMMs use V_WMMA_F32_16X16X4_F32 (fp32 in/out, matches fp32 reference).
//
// WMMA f32 16x16x4 layout assumptions (per CDNA5 ISA 7.12.2):
//   A (16x4, MxK): lane 0-15 -> M=lane, vgpr0/1 = K0/K1; lane 16-31 -> K2/K3
//   B (4x16, KxN): lane 0-15 -> N=lane, vgpr0/1 = K0/K1; lane 16-31 -> K2/K3
//   C/D (16x16)  : vgpr r, lanes 0-15: M=r, N=lane; lanes 16-31: M=r+8, N=lane-16
// ---------------------------------------------------------------------------

typedef __attribute__((ext_vector_type(2))) float v2f;
typedef __attribute__((ext_vector_type(8))) float v8f;

__device__ __forceinline__ v8f wmma4(v2f a, v2f b, v8f c) {
  // (neg_a, A, neg_b, B, c_mod, C, reuse_a, reuse_b)
  return __builtin_amdgcn_wmma_f32_16x16x4_f32(false, a, false, b, (short)0, c,
                                               false, false);
}

__device__ __forceinline__ v8f vzero8() {
  v8f z;
#pragma unroll
  for (int i = 0; i < 8; ++i) z[i] = 0.f;
  return z;
}

// order-preserving map: larger float -> larger uint
__device__ __forceinline__ unsigned fkey(float f) {
  unsigned u = __float_as_uint(f);
  return (u & 0x80000000u) ? ~u : (u | 0x80000000u);
}

__device__ __forceinline__ int wred_add(int v) {
#pragma unroll
  for (int o = 16; o; o >>= 1) v += __shfl_xor(v, o, 32);
  return v;
}
__device__ __forceinline__ float wred_addf(float v) {
#pragma unroll
  for (int o = 16; o; o >>= 1) v += __shfl_xor(v, o, 32);
  return v;
}
__device__ __forceinline__ float wred_maxf(float v) {
#pragma unroll
  for (int o = 16; o; o >>= 1) { float t = __shfl_xor(v, o, 32); if (t > v) v = t; }
  return v;
}

// ---------------------------------------------------------------------------
// Kernel A: window-gather + LayerNorm + 128x128 projection.
// mode 0: q (N=1536, 16x16 windows); mode 1: k/v (N=384, 8x8 windows).
// Output layout: out[((h*36 + l)*N + i)*32 + d],  feature j = h*32+d.
// ---------------------------------------------------------------------------
__global__ __launch_bounds__(128) void ln_proj_kernel(
    const float* __restrict__ X, const float* __restrict__ gamma,
    const float* __restrict__ beta, const float* __restrict__ W,
    const float* __restrict__ bias, float* __restrict__ outp, int N, int mode) {
  __shared__ float xs[16][129];
  const int tid = threadIdx.x;
  const int tt = tid >> 3, sub = tid & 7;
  const int gt = blockIdx.x * 16 + tt;
  const int l = gt / N, i = gt % N;
  const int x = l / 6, y = l % 6;
  long src;
  if (mode == 0) {
    int n = i >> 8, w1 = (i >> 4) & 15, w2 = i & 15;
    src = ((((long)(n * 6 + x) * 6 + y) * 16 + w1) * 16 + w2) * 128;
  } else {
    int n = i >> 6, w1 = (i >> 3) & 7, w2 = i & 7;
    src = ((((long)(n * 6 + x) * 6 + y) * 8 + w1) * 8 + w2) * 128;
  }
  const float* xp = X + src;
  float vals[16];
  float s = 0.f, ss = 0.f;
#pragma unroll
  for (int j = 0; j < 16; ++j) {
    float v = xp[sub * 16 + j];
    vals[j] = v; s += v; ss += v * v;
  }
#pragma unroll
  for (int o = 1; o < 8; o <<= 1) { s += __shfl_xor(s, o, 32); ss += __shfl_xor(ss, o, 32); }
  float mu = s * (1.f / 128.f);
  float var = ss * (1.f / 128.f) - mu * mu;
  float rstd = rsqrtf(var + 1e-5f);
#pragma unroll
  for (int j = 0; j < 16; ++j) {
    int f = sub * 16 + j;
    xs[tt][f] = (vals[j] - mu) * rstd * gamma[f] + beta[f];
  }
  __syncthreads();

  const int w = tid >> 5, lane = tid & 31;
  const int m = lane & 15, kh2 = (lane >> 4) * 2;
#pragma unroll
  for (int p = 0; p < 2; ++p) {
    int nt = w + p * 4;
    int ncol = nt * 16 + m;
    v8f acc = vzero8();
    for (int kk = 0; kk < 32; ++kk) {
      int k0 = kk * 4 + kh2;
      v2f a; a[0] = xs[m][k0]; a[1] = xs[m][k0 + 1];
      v2f bb; bb[0] = W[k0 * 128 + ncol]; bb[1] = W[(k0 + 1) * 128 + ncol];
      acc = wmma4(a, bb, acc);
    }
    int j = ncol, h = j >> 5, d = j & 31;
    float bj = bias[j];
#pragma unroll
    for (int r = 0; r < 8; ++r) {
      int M = r + (lane >> 4) * 8;
      int g2 = blockIdx.x * 16 + M;
      int l2 = g2 / N, i2 = g2 % N;
      outp[((size_t)(h * 36 + l2) * N + i2) * 32 + d] = acc[r] + bj;
    }
  }
}

// ---------------------------------------------------------------------------
// Kernel B: query pruning. One wave per (h,l) row of 1536 queries.
// Exact top-1152 by squared-norm saliency (radix select, lowest-index ties).
// Pruned query rows of qh are zeroed in place.
// ---------------------------------------------------------------------------
__global__ __launch_bounds__(32) void prune_kernel(float* __restrict__ qh) {
  __shared__ float sal[1536];
  const int lane = threadIdx.x;
  float* base = qh + (size_t)blockIdx.x * 1536 * 32;
  for (int t = lane; t < 1536; t += 32) {
    const float* p = base + t * 32;
    float s = 0.f;
#pragma unroll
    for (int d = 0; d < 32; ++d) { float v = p[d]; s += v * v; }
    sal[t] = s;
  }
  __syncthreads();
  // radix select: 1152-th largest
  unsigned prefix = 0;
  int k = 1152;
  for (int bit = 31; bit >= 0; --bit) {
    unsigned want = (prefix | (1u << bit)) >> bit;
    int cnt = 0;
    for (int t = lane; t < 1536; t += 32)
      if ((fkey(sal[t]) >> bit) == want) cnt++;
    cnt = wred_add(cnt);
    if (cnt >= k) prefix |= (1u << bit); else k -= cnt;
  }
  const unsigned thrU = prefix;
  int G = 0;
  for (int t = lane; t < 1536; t += 32)
    if (fkey(sal[t]) > thrU) G++;
  G = wred_add(G);
  const int keepEq = 1152 - G;
  for (int t = lane; t < 1536; t += 32) {
    unsigned u = fkey(sal[t]);
    bool keep;
    if (u > thrU) keep = true;
    else if (u < thrU) keep = false;
    else {
      int rank = 0;
      for (int j = 0; j < t; ++j)
        if (fkey(sal[j]) == thrU) rank++;
      keep = rank < keepEq;
    }
    if (!keep) {
      float* p = base + t * 32;
#pragma unroll
      for (int d = 0; d < 32; ++d) p[d] = 0.f;
    }
  }
}

// ---------------------------------------------------------------------------
// Kernel C: attention. grid (144 = h*l, 6 q-chunks), 256 threads (8 waves).
// Per 16-query tile: logits (WMMA), +bias, exact top-96 mask, softmax, att@V.
// ---------------------------------------------------------------------------
__global__ __launch_bounds__(256) void attn_kernel(
    const float* __restrict__ qh, const float* __restrict__ kh,
    const float* __restrict__ vh, const float* __restrict__ logit_bias,
    float* __restrict__ aout) {
  __shared__ float qa[16][33];
  __shared__ float lg[16][388];
  __shared__ float part[4][2][256];

  const int hl = blockIdx.x;            // h*36 + l
  const int qbase = blockIdx.y * 256;
  const int tid = threadIdx.x, w = tid >> 5, lane = tid & 31;
  const int m = lane & 15, kh2 = (lane >> 4) * 2;
  const float* qhp = qh + (size_t)hl * 1536 * 32;
  const float* khp = kh + (size_t)hl * 384 * 32;
  const float* vhp = vh + (size_t)hl * 384 * 32;

  float lb = 0.f;
#pragma unroll
  for (int c = 0; c < 6; ++c) lb += logit_bias[c];
  lb *= (1.f / 6.f);
  const float scale = 0.17677669529663687f;  // 32^-0.5

  for (int qt = 0; qt < 16; ++qt) {
    const int q0 = qbase + qt * 16;
    for (int idx = tid; idx < 512; idx += 256) {
      int mm = idx >> 5, d = idx & 31;
      qa[mm][d] = qhp[(size_t)(q0 + mm) * 32 + d] * scale;
    }
    __syncthreads();

    // logits 16x384: 24 n-tiles, 3 per wave
#pragma unroll
    for (int p = 0; p < 3; ++p) {
      int nt = w * 3 + p;
      int key = nt * 16 + m;
      v8f acc = vzero8();
#pragma unroll
      for (int kk = 0; kk < 8; ++kk) {
        int k0 = kk * 4 + kh2;
        v2f a; a[0] = qa[m][k0]; a[1] = qa[m][k0 + 1];
        v2f bb; bb[0] = khp[key * 32 + k0]; bb[1] = khp[key * 32 + k0 + 1];
        acc = wmma4(a, bb, acc);
      }
#pragma unroll
      for (int r = 0; r < 8; ++r) {
        int M = r + (lane >> 4) * 8;
        lg[M][nt * 16 + m] = acc[r] + lb;
      }
    }
    __syncthreads();

    // exact top-96 + softmax, 2 rows per wave
#pragma unroll
    for (int rr = 0; rr < 2; ++rr) {
      int row = w + rr * 8;
      unsigned prefix = 0;
      int k = 96;
      for (int bit = 31; bit >= 0; --bit) {
        unsigned want = (prefix | (1u << bit)) >> bit;
        int cnt = 0;
#pragma unroll
        for (int jj = 0; jj < 12; ++jj)
          if ((fkey(lg[row][lane + jj * 32]) >> bit) == want) cnt++;
        cnt = wred_add(cnt);
        if (cnt >= k) prefix |= (1u << bit); else k -= cnt;
      }
      const unsigned thrU = prefix;
      int G = 0;
#pragma unroll
      for (int jj = 0; jj < 12; ++jj)
        if (fkey(lg[row][lane + jj * 32]) > thrU) G++;
      G = wred_add(G);
      const int keepEq = 96 - G;
      float vals[12]; bool keeps[12];
      float mx = -3.0e38f;
#pragma unroll
      for (int jj = 0; jj < 12; ++jj) {
        int c = lane + jj * 32;
        float v = lg[row][c];
        unsigned u = fkey(v);
        bool keep;
        if (u > thrU) keep = true;
        else if (u < thrU) keep = false;
        else {
          int rank = 0;
          for (int j2 = 0; j2 < c; ++j2)
            if (fkey(lg[row][j2]) == thrU) rank++;
          keep = rank < keepEq;   // lowest-index tie break, like lax.top_k
        }
        keeps[jj] = keep; vals[jj] = v;
        if (keep && v > mx) mx = v;
      }
      mx = wred_maxf(mx);
      float sum = 0.f, ex[12];
#pragma unroll
      for (int jj = 0; jj < 12; ++jj) {
        float e = keeps[jj] ? expf(vals[jj] - mx) : 0.f;
        ex[jj] = e; sum += e;
      }
      sum = wred_addf(sum);
      float inv = 1.f / sum;
#pragma unroll
      for (int jj = 0; jj < 12; ++jj) lg[row][lane + jj * 32] = ex[jj] * inv;
    }
    __syncthreads();

    // att(16x384) @ vh(384x32): 2 n-tiles x 4 K-quarters across 8 waves
    {
      int nt = w & 1, kq = w >> 1;
      int d = nt * 16 + m;
      v8f acc = vzero8();
      for (int kk = kq * 24; kk < kq * 24 + 24; ++kk) {
        int k0 = kk * 4 + kh2;
        v2f a; a[0] = lg[m][k0]; a[1] = lg[m][k0 + 1];
        v2f bb; bb[0] = vhp[(size_t)k0 * 32 + d]; bb[1] = vhp[(size_t)(k0 + 1) * 32 + d];
        acc = wmma4(a, bb, acc);
      }
#pragma unroll
      for (int r = 0; r < 8; ++r) part[kq][nt][r * 32 + lane] = acc[r];
    }
    __syncthreads();
    if (tid < 64) {
      int nt = w;  // 0 or 1
#pragma unroll
      for (int r = 0; r < 8; ++r) {
        float s = part[0][nt][r * 32 + lane] + part[1][nt][r * 32 + lane] +
                  part[2][nt][r * 32 + lane] + part[3][nt][r * 32 + lane];
        int M = r + (lane >> 4) * 8;
        int d = nt * 16 + (lane & 15);
        aout[((size_t)hl * 1536 + q0 + M) * 32 + d] = s;
      }
    }
    __syncthreads();
  }
}

// ---------------------------------------------------------------------------
// Kernel D: output projection + camera mean + skip. grid 576, 128 threads.
// ---------------------------------------------------------------------------
__global__ __launch_bounds__(128) void outproj_kernel(
    const float* __restrict__ aout, const float* __restrict__ Wp,
    const float* __restrict__ bp, const float* __restrict__ skip,
    float* __restrict__ z) {
  __shared__ float as[16][129];
  const int tid = threadIdx.x, w = tid >> 5, lane = tid & 31;
  const int m = lane & 15, kh2 = (lane >> 4) * 2;
  const int t0 = blockIdx.x * 16;
  const int l = t0 >> 8;
  const int w1 = (t0 & 255) >> 4;

  v8f acc[2];
  acc[0] = vzero8(); acc[1] = vzero8();

  for (int cam = 0; cam < 6; ++cam) {
    for (int idx = tid; idx < 2048; idx += 128) {
      int tok = idx >> 7, j = idx & 127;
      int h = j >> 5, d = j & 31;
      int qi = cam * 256 + w1 * 16 + tok;
      as[tok][j] = aout[((size_t)(h * 36 + l) * 1536 + qi) * 32 + d];
    }
    __syncthreads();
#pragma unroll
    for (int p = 0; p < 2; ++p) {
      int nt = w + p * 4;
      int ncol = nt * 16 + m;
      for (int kk = 0; kk < 32; ++kk) {
        int k0 = kk * 4 + kh2;
        v2f a; a[0] = as[m][k0]; a[1] = as[m][k0 + 1];
        v2f bb; bb[0] = Wp[k0 * 128 + ncol]; bb[1] = Wp[(k0 + 1) * 128 + ncol];
        acc[p] = wmma4(a, bb, acc[p]);
      }
    }
    __syncthreads();
  }
#pragma unroll
  for (int p = 0; p < 2; ++p) {
    int nt = w + p * 4;
    int j = nt * 16 + (lane & 15);
    float bj = bp[j];
#pragma unroll
    for (int r = 0; r < 8; ++r) {
      int M = r + (lane >> 4) * 8;
      size_t tg = (size_t)(t0 + M);
      z[tg * 128 + j] = acc[p][r] * (1.f / 6.f) + bj + skip[tg * 128 + j];
    }
  }
}

// ---------------------------------------------------------------------------
extern "C" void kernel_launch(void* const* d_in, const int* in_sizes, int n_in,
                              void* d_out, int out_size, void* d_ws,
                              size_t ws_size, hipStream_t stream) {
  const float* q   = (const float*)d_in[0];
  const float* kin = (const float*)d_in[1];
  const float* vin = (const float*)d_in[2];
  const float* skip = (const float*)d_in[3];
  const float* logit_bias = (const float*)d_in[4];
  const float* g_q = (const float*)d_in[5];
  const float* b_q = (const float*)d_in[6];
  const float* g_k = (const float*)d_in[7];
  const float* b_k = (const float*)d_in[8];
  const float* g_v = (const float*)d_in[9];
  const float* b_v = (const float*)d_in[10];
  const float* Wq = (const float*)d_in[11];
  const float* bq = (const float*)d_in[12];
  const float* Wk = (const float*)d_in[13];
  const float* bk = (const float*)d_in[14];
  const float* Wv = (const float*)d_in[15];
  const float* bv = (const float*)d_in[16];
  const float* Wp = (const float*)d_in[17];
  const float* bp = (const float*)d_in[18];

  float* ws = (float*)d_ws;
  float* qh = ws;                       // 4*36*1536*32 = 7,077,888 floats
  float* kh = qh + 7077888;             // 4*36*384*32  = 1,769,472 floats
  float* vh = kh + 1769472;             // 1,769,472 floats
  float* aout = vh + 1769472;           // 7,077,888 floats

  ln_proj_kernel<<<55296 / 16, 128, 0, stream>>>(q, g_q, b_q, Wq, bq, qh, 1536, 0);
  ln_proj_kernel<<<13824 / 16, 128, 0, stream>>>(kin, g_k, b_k, Wk, bk, kh, 384, 1);
  ln_proj_kernel<<<13824 / 16, 128, 0, stream>>>(vin, g_v, b_v, Wv, bv, vh, 384, 1);
  prune_kernel<<<144, 32, 0, stream>>>(qh);
  dim3 gc(144, 6);
  attn_kernel<<<gc, 256, 0, stream>>>(qh, kh, vh, logit_bias, aout);
  outproj_kernel<<<576, 128, 0, stream>>>(aout, Wp, bp, skip, (float*)d_out);
}